// GINModel_53901839564968
// MI455X (gfx1250) — compile-verified
//
#include <hip/hip_runtime.h>

typedef __attribute__((ext_vector_type(2))) float v2f;
typedef __attribute__((ext_vector_type(8))) float v8f;

#define N_NODES 50000
#define N_EDGES 800000
#define HID 64
#define PLANE (N_NODES * HID) /* 3,200,000 floats per stored layer output */

// ---------------------------------------------------------------------------
// Zero-fill scratch (agg accumulator / pooled graph buffer)
// ---------------------------------------------------------------------------
__global__ void fill_zero_kernel(float* __restrict__ p, long n) {
  long i = (long)blockIdx.x * blockDim.x + threadIdx.x;
  if (i < n) p[i] = 0.0f;
}

// ---------------------------------------------------------------------------
// agg[dst[e]][c] += h[src[e]][c]  (one thread per edge-channel, fp32 atomics)
// ---------------------------------------------------------------------------
__global__ void scatter_add_kernel(const float* __restrict__ hin,
                                   const int* __restrict__ src,
                                   const int* __restrict__ dst,
                                   float* __restrict__ agg, int dLog2) {
  long tid = (long)blockIdx.x * blockDim.x + threadIdx.x;
  long total = (long)N_EDGES << dLog2;
  if (tid >= total) return;
  int c = (int)(tid & ((1L << dLog2) - 1));
  int e = (int)(tid >> dLog2);
  int s = src[e];
  int t = dst[e];
  float v = hin[((size_t)s << dLog2) + c];
  atomicAdd(&agg[((size_t)t << dLog2) + c], v);
}

// ---------------------------------------------------------------------------
// GIN layer: out = relu( relu((agg+h) @ W1 + b1) @ W2 + b2 )
// One wave computes one 16x64 row tile; 8 waves per block.
// FP32 WMMA: V_WMMA_F32_16X16X4_F32 (exact fp32, matrix path not bottleneck).
// A-frag (16x4 f32): lane L holds A[L&15][k0+2*(L>>4) + {0,1}] in {a.x,a.y}
// B-frag (4x16 f32): lane L holds B[k0+2*(L>>4) + {0,1}][L&15] in {b.x,b.y}
// C/D (16x16 f32):   elem v of lane L = D[v + 8*(L>>4)][L&15 + 16*nt]
// ---------------------------------------------------------------------------
__global__ __launch_bounds__(256) void gin_layer_kernel(
    const float* __restrict__ agg, const float* __restrict__ hin, int K,
    const float* __restrict__ w1, const float* __restrict__ bias1,
    const float* __restrict__ w2, const float* __restrict__ bias2,
    float* __restrict__ outp) {
  __shared__ float t_lds[8][16][66];  // padded stride 66 -> no bank conflicts

  const int lane = threadIdx.x & 31;
  const int wv = threadIdx.x >> 5;
  const int mrow = lane & 15;
  const int khalf = lane >> 4;  // 0 for lanes 0-15, 1 for lanes 16-31

  const int tile = blockIdx.x * 8 + wv;
  const bool active = (tile * 16) < N_NODES;
  const int rowbase = active ? tile * 16 : (N_NODES - 16);  // clamp (uniform)

  // ---- stage 1: t = relu((agg+h) @ W1 + b1), K = 128 or 64 ----
  v8f acc0 = {}, acc1 = {}, acc2 = {}, acc3 = {};
  for (int k0 = 0; k0 < K; k0 += 4) {
    const int kk = k0 + 2 * khalf;
    const size_t ai = (size_t)(rowbase + mrow) * K + kk;
    v2f a;
    a.x = agg[ai] + hin[ai];
    a.y = agg[ai + 1] + hin[ai + 1];

    const float* pw = w1 + (size_t)kk * HID + mrow;
    v2f b0, b1, b2, b3;
    b0.x = pw[0];  b0.y = pw[HID + 0];
    b1.x = pw[16]; b1.y = pw[HID + 16];
    b2.x = pw[32]; b2.y = pw[HID + 32];
    b3.x = pw[48]; b3.y = pw[HID + 48];

    acc0 = __builtin_amdgcn_wmma_f32_16x16x4_f32(false, a, false, b0, (short)0, acc0, false, false);
    acc1 = __builtin_amdgcn_wmma_f32_16x16x4_f32(false, a, false, b1, (short)0, acc1, false, false);
    acc2 = __builtin_amdgcn_wmma_f32_16x16x4_f32(false, a, false, b2, (short)0, acc2, false, false);
    acc3 = __builtin_amdgcn_wmma_f32_16x16x4_f32(false, a, false, b3, (short)0, acc3, false, false);
  }

  // bias + relu, stage into LDS in row-major order for stage-2 A fragments
#pragma unroll
  for (int v = 0; v < 8; ++v) {
    const int r = v + 8 * khalf;
    t_lds[wv][r][0 + mrow]  = fmaxf(acc0[v] + bias1[0 + mrow], 0.0f);
    t_lds[wv][r][16 + mrow] = fmaxf(acc1[v] + bias1[16 + mrow], 0.0f);
    t_lds[wv][r][32 + mrow] = fmaxf(acc2[v] + bias1[32 + mrow], 0.0f);
    t_lds[wv][r][48 + mrow] = fmaxf(acc3[v] + bias1[48 + mrow], 0.0f);
  }
  __syncthreads();

  // ---- stage 2: out = relu(t @ W2 + b2), K = 64 ----
  v8f d0 = {}, d1 = {}, d2 = {}, d3 = {};
  for (int k0 = 0; k0 < HID; k0 += 4) {
    const int kk = k0 + 2 * khalf;
    v2f a;
    a.x = t_lds[wv][mrow][kk];
    a.y = t_lds[wv][mrow][kk + 1];

    const float* pw = w2 + (size_t)kk * HID + mrow;
    v2f b0, b1, b2, b3;
    b0.x = pw[0];  b0.y = pw[HID + 0];
    b1.x = pw[16]; b1.y = pw[HID + 16];
    b2.x = pw[32]; b2.y = pw[HID + 32];
    b3.x = pw[48]; b3.y = pw[HID + 48];

    d0 = __builtin_amdgcn_wmma_f32_16x16x4_f32(false, a, false, b0, (short)0, d0, false, false);
    d1 = __builtin_amdgcn_wmma_f32_16x16x4_f32(false, a, false, b1, (short)0, d1, false, false);
    d2 = __builtin_amdgcn_wmma_f32_16x16x4_f32(false, a, false, b2, (short)0, d2, false, false);
    d3 = __builtin_amdgcn_wmma_f32_16x16x4_f32(false, a, false, b3, (short)0, d3, false, false);
  }

  if (active) {
#pragma unroll
    for (int v = 0; v < 8; ++v) {
      float* po = outp + (size_t)(rowbase + v + 8 * khalf) * HID;
      po[0 + mrow]  = fmaxf(d0[v] + bias2[0 + mrow], 0.0f);
      po[16 + mrow] = fmaxf(d1[v] + bias2[16 + mrow], 0.0f);
      po[32 + mrow] = fmaxf(d2[v] + bias2[32 + mrow], 0.0f);
      po[48 + mrow] = fmaxf(d3[v] + bias2[48 + mrow], 0.0f);
    }
  }
}

// ---------------------------------------------------------------------------
// JK projection: hjk = concat(xs[0..4], axis=1) @ jk_w + jk_b   (K = 320)
// ---------------------------------------------------------------------------
__global__ __launch_bounds__(256) void jk_kernel(const float* __restrict__ xs,
                                                 const float* __restrict__ jkw,
                                                 const float* __restrict__ jkb,
                                                 float* __restrict__ hjk) {
  const int lane = threadIdx.x & 31;
  const int wv = threadIdx.x >> 5;
  const int mrow = lane & 15;
  const int khalf = lane >> 4;

  const int tile = blockIdx.x * 8 + wv;
  if (tile * 16 >= N_NODES) return;  // no LDS/barrier here, early-exit OK
  const int rowbase = tile * 16;

  v8f acc0 = {}, acc1 = {}, acc2 = {}, acc3 = {};
  for (int k0 = 0; k0 < 5 * HID; k0 += 4) {
    const int kk = k0 + 2 * khalf;
    const int plane = kk >> 6;   // which layer output
    const int c = kk & 63;       // kk,kk+1 stay inside one plane (kk even)
    const size_t ai = (size_t)plane * PLANE + (size_t)(rowbase + mrow) * HID + c;
    v2f a;
    a.x = xs[ai];
    a.y = xs[ai + 1];

    const float* pw = jkw + (size_t)kk * HID + mrow;
    v2f b0, b1, b2, b3;
    b0.x = pw[0];  b0.y = pw[HID + 0];
    b1.x = pw[16]; b1.y = pw[HID + 16];
    b2.x = pw[32]; b2.y = pw[HID + 32];
    b3.x = pw[48]; b3.y = pw[HID + 48];

    acc0 = __builtin_amdgcn_wmma_f32_16x16x4_f32(false, a, false, b0, (short)0, acc0, false, false);
    acc1 = __builtin_amdgcn_wmma_f32_16x16x4_f32(false, a, false, b1, (short)0, acc1, false, false);
    acc2 = __builtin_amdgcn_wmma_f32_16x16x4_f32(false, a, false, b2, (short)0, acc2, false, false);
    acc3 = __builtin_amdgcn_wmma_f32_16x16x4_f32(false, a, false, b3, (short)0, acc3, false, false);
  }

#pragma unroll
  for (int v = 0; v < 8; ++v) {
    float* po = hjk + (size_t)(rowbase + v + 8 * khalf) * HID;
    po[0 + mrow]  = acc0[v] + jkb[0 + mrow];
    po[16 + mrow] = acc1[v] + jkb[16 + mrow];
    po[32 + mrow] = acc2[v] + jkb[32 + mrow];
    po[48 + mrow] = acc3[v] + jkb[48 + mrow];
  }
}

// ---------------------------------------------------------------------------
// global_add_pool: g[batch[n]][c] += hjk[n][c]
// ---------------------------------------------------------------------------
__global__ void pool_kernel(const float* __restrict__ hjk,
                            const int* __restrict__ batch,
                            float* __restrict__ g) {
  long tid = (long)blockIdx.x * blockDim.x + threadIdx.x;
  if (tid >= (long)PLANE) return;
  int n = (int)(tid >> 6);
  int c = (int)(tid & 63);
  atomicAdd(&g[(size_t)batch[n] * HID + c], hjk[tid]);
}

// ---------------------------------------------------------------------------
// Classifier: t = g@W1+b1; batchnorm(batch stats); relu; out = t@W2+b2
// Tiny (128x64, 128x10): single block, scalar math, LDS staging.
// ---------------------------------------------------------------------------
__global__ __launch_bounds__(256) void classifier_kernel(
    const float* __restrict__ g, const float* __restrict__ w1,
    const float* __restrict__ b1, const float* __restrict__ gamma,
    const float* __restrict__ beta, const float* __restrict__ w2,
    const float* __restrict__ b2, float* __restrict__ out) {
  __shared__ float t[128][HID];
  __shared__ float mu[HID];
  __shared__ float rstd[HID];
  const int tid = threadIdx.x;

  for (int idx = tid; idx < 128 * HID; idx += 256) {
    int i = idx >> 6, j = idx & 63;
    float s = b1[j];
    for (int k = 0; k < HID; ++k) s += g[i * HID + k] * w1[k * HID + j];
    t[i][j] = s;
  }
  __syncthreads();

  if (tid < HID) {
    float s = 0.0f, ss = 0.0f;
    for (int i = 0; i < 128; ++i) {
      float v = t[i][tid];
      s += v;
      ss += v * v;
    }
    float m = s * (1.0f / 128.0f);
    float var = ss * (1.0f / 128.0f) - m * m;
    mu[tid] = m;
    rstd[tid] = rsqrtf(var + 1e-5f);
  }
  __syncthreads();

  for (int idx = tid; idx < 128 * HID; idx += 256) {
    int i = idx >> 6, j = idx & 63;
    float v = (t[i][j] - mu[j]) * rstd[j] * gamma[j] + beta[j];
    t[i][j] = fmaxf(v, 0.0f);
  }
  __syncthreads();

  for (int idx = tid; idx < 128 * 10; idx += 256) {
    int i = idx / 10, j = idx % 10;
    float s = b2[j];
    for (int k = 0; k < HID; ++k) s += t[i][k] * w2[k * 10 + j];
    out[idx] = s;
  }
}

// ---------------------------------------------------------------------------
extern "C" void kernel_launch(void* const* d_in, const int* in_sizes, int n_in,
                              void* d_out, int out_size, void* d_ws, size_t ws_size,
                              hipStream_t stream) {
  (void)in_sizes; (void)n_in; (void)out_size; (void)ws_size;

  const float* x        = (const float*)d_in[0];
  const int*   ei       = (const int*)d_in[1];
  const int*   batch    = (const int*)d_in[2];
  const float* w1_0     = (const float*)d_in[3];
  const float* b1_0     = (const float*)d_in[4];
  const float* w2_0     = (const float*)d_in[5];
  const float* b2_0     = (const float*)d_in[6];
  const float* w1_rest  = (const float*)d_in[7];
  const float* b1_rest  = (const float*)d_in[8];
  const float* w2_rest  = (const float*)d_in[9];
  const float* b2_rest  = (const float*)d_in[10];
  const float* jk_w     = (const float*)d_in[11];
  const float* jk_b     = (const float*)d_in[12];
  const float* clf_w1   = (const float*)d_in[13];
  const float* clf_b1   = (const float*)d_in[14];
  const float* bn_gamma = (const float*)d_in[15];
  const float* bn_beta  = (const float*)d_in[16];
  const float* clf_w2   = (const float*)d_in[17];
  const float* clf_b2   = (const float*)d_in[18];

  const int* src = ei;
  const int* dst = ei + N_EDGES;

  // workspace layout (floats): agg[50000*128] | xs[5][50000*64] | hjk | g
  float* ws  = (float*)d_ws;
  float* agg = ws;
  float* xs  = ws + (size_t)N_NODES * 128;
  float* hjk = xs + 5 * (size_t)PLANE;
  float* g   = hjk + (size_t)PLANE;

  const unsigned GEMM_BLOCKS = (N_NODES / 16 + 7) / 8;  // 391

  for (int i = 0; i < 5; ++i) {
    const int dlog = (i == 0) ? 7 : 6;
    const int d = 1 << dlog;
    const float* hin = (i == 0) ? x : (xs + (size_t)(i - 1) * PLANE);

    long na = (long)N_NODES * d;
    fill_zero_kernel<<<(unsigned)((na + 255) / 256), 256, 0, stream>>>(agg, na);

    long te = (long)N_EDGES * d;
    scatter_add_kernel<<<(unsigned)((te + 255) / 256), 256, 0, stream>>>(hin, src, dst, agg, dlog);

    const float *w1, *bb1, *w2, *bb2;
    if (i == 0) {
      w1 = w1_0; bb1 = b1_0; w2 = w2_0; bb2 = b2_0;
    } else {
      w1  = w1_rest + (size_t)(i - 1) * HID * HID;
      bb1 = b1_rest + (size_t)(i - 1) * HID;
      w2  = w2_rest + (size_t)(i - 1) * HID * HID;
      bb2 = b2_rest + (size_t)(i - 1) * HID;
    }
    gin_layer_kernel<<<GEMM_BLOCKS, 256, 0, stream>>>(
        agg, hin, d, w1, bb1, w2, bb2, xs + (size_t)i * PLANE);
  }

  jk_kernel<<<GEMM_BLOCKS, 256, 0, stream>>>(xs, jk_w, jk_b, hjk);

  fill_zero_kernel<<<(128 * HID + 255) / 256, 256, 0, stream>>>(g, 128 * HID);
  pool_kernel<<<(unsigned)(((long)PLANE + 255) / 256), 256, 0, stream>>>(hjk, batch, g);

  classifier_kernel<<<1, 256, 0, stream>>>(g, clf_w1, clf_b1, bn_gamma, bn_beta,
                                           clf_w2, clf_b2, (float*)d_out);
}